// SignedMagnet_67585605369884
// MI455X (gfx1250) — compile-verified
//
#include <hip/hip_runtime.h>
#include <math.h>

#define N_NODES 50000
#define N_EDGES 800000
#define IN_DIM  128
#define HID     64
#define OUTD    8
#define NTILES  (N_NODES / 16)   // 3125 exactly
#define TPAD    66               // padded LDS tile row stride (bank-conflict-free b64)

typedef __attribute__((ext_vector_type(2))) float v2f;
typedef __attribute__((ext_vector_type(4))) float v4f;
typedef __attribute__((ext_vector_type(8))) float v8f;

__device__ __forceinline__ v8f wmma_f32_16x16x4(v2f a, v2f b, v8f c) {
  // D = A(16x4,f32) * B(4x16,f32) + C(16x16,f32); 8-arg VOP3P form
  return __builtin_amdgcn_wmma_f32_16x16x4_f32(false, a, false, b, (short)0, c,
                                               false, false);
}

// ---------------------------------------------------------------------------
// Kernel 1: input transform  out = relu(h @ W^T + b),  h:[N,128] W:[64,128]
// One wave computes a [16 x 64] output tile via WMMA f32 16x16x4, K=128.
// Weights staged in LDS in fragment-paired layout:
//   sWp[(k>>1)*2*HID + 2*j + (k&1)] = W[j][k]
// so each B fragment is one aligned ds_load_b64 straight into a VGPR pair
// (no repack moves), consecutive lanes -> distinct bank pairs.
// ---------------------------------------------------------------------------
__global__ __launch_bounds__(128) void k_transform(
    const float* __restrict__ h, const float* __restrict__ W,
    const float* __restrict__ bias, float* __restrict__ out) {
  __shared__ float sWp[IN_DIM * HID];  // 32 KB, pair layout
  const int tid = threadIdx.x;
  for (int it = 0; it < (IN_DIM * HID) / 128; ++it) {
    int flat = it * 128 + tid;
    int j = flat >> 7;      // output channel
    int k = flat & 127;     // input channel
    sWp[(k >> 1) * (2 * HID) + 2 * j + (k & 1)] = W[flat];
  }
  __syncthreads();

  const int lane = tid & 31;
  const int wave = tid >> 5;
  const int half = lane >> 4;
  const int l    = lane & 15;
  const int tile = blockIdx.x * 4 + wave;
  if (tile >= NTILES) return;            // wave-uniform: EXEC all-ones below
  const int row0 = tile * 16;

  v8f acc[4];
  for (int jt = 0; jt < 4; ++jt) {
    float bv = bias[jt * 16 + l];
    for (int i = 0; i < 8; ++i) acc[jt][i] = bv;
  }

  const float* hrow = h + (size_t)(row0 + l) * IN_DIM + 2 * half;
  for (int k0 = 0; k0 < IN_DIM; k0 += 4) {
    v2f a = *(const v2f*)(hrow + k0);              // A[l][k0+2h], A[l][k0+2h+1]
    const float* wrow = sWp + ((k0 >> 1) + half) * (2 * HID);
    for (int jt = 0; jt < 4; ++jt) {
      v2f b = *(const v2f*)(wrow + 2 * (jt * 16 + l));
      acc[jt] = wmma_f32_16x16x4(a, b, acc[jt]);
    }
  }

  for (int jt = 0; jt < 4; ++jt)
    for (int i = 0; i < 8; ++i) {
      float v = acc[jt][i];
      v = v > 0.f ? v : 0.f;
      out[(size_t)(row0 + i + 8 * half) * HID + jt * 16 + l] = v;
    }
}

// ---------------------------------------------------------------------------
// Kernel 2: zero scratch
// ---------------------------------------------------------------------------
__global__ void k_zero(float* __restrict__ p, int n) {
  int i = blockIdx.x * blockDim.x + threadIdx.x;
  const int stride = gridDim.x * blockDim.x;
  for (; i < n; i += stride) p[i] = 0.f;
}

// ---------------------------------------------------------------------------
// Kernel 3: fused complex edge aggregation (gather * e, scatter-add to dst).
// One wave per edge; lane covers 2 of the 64 features (coalesced b64 loads,
// f32 atomic adds resolving in L2 — z buffers are L2-resident at 12.8 MB).
// ---------------------------------------------------------------------------
__global__ __launch_bounds__(256) void k_agg(
    const float* __restrict__ hr, const float* __restrict__ hi,
    const float* __restrict__ deg, const float* __restrict__ wre,
    const float* __restrict__ wim, const int* __restrict__ src,
    const int* __restrict__ dst, float* __restrict__ zr,
    float* __restrict__ zi) {
  const int lane = threadIdx.x & 31;
  const long long gw =
      (long long)blockIdx.x * (blockDim.x >> 5) + (threadIdx.x >> 5);
  if (gw >= N_EDGES) return;
  const int e = (int)gw;
  const int s = src[e];
  const int t = dst[e];
  const float coef = deg[t] * deg[s];
  const float er = coef * wre[e];
  const float ei = coef * wim[e];
  const int f = lane * 2;
  v2f xr = *(const v2f*)(hr + (size_t)s * HID + f);
  v2f xi = *(const v2f*)(hi + (size_t)s * HID + f);
  float* pr = zr + (size_t)t * HID + f;
  float* pi = zi + (size_t)t * HID + f;
  atomicAdd(pr + 0, er * xr.x - ei * xi.x);
  atomicAdd(pr + 1, er * xr.y - ei * xi.y);
  atomicAdd(pi + 0, ei * xr.x + er * xi.x);
  atomicAdd(pi + 1, ei * xr.y + er * xi.y);
}

// ---------------------------------------------------------------------------
// Kernel 4: dense complex layer (WMMA).
//   P = zr@W1^T + b1 ; Q = zi@W2^T + b2 ; zr_new = P - Q
//   R = zr_new@W2^T + b2 ; S = zi@W1^T + b1 ; zi_new = R + S
//   hr = relu(zr_new) ; hi = relu(zi_new)
// Weights in fragment-paired LDS layout (single b64 per B fragment).
// zr_new D-fragments are round-tripped through a per-wave LDS tile to be
// re-striped into A-fragment layout (same-wave LDS ordering, no barrier).
// 2 waves / block -> LDS = 32 KB weights + 16.9 KB tiles.
// ---------------------------------------------------------------------------
__global__ __launch_bounds__(64) void k_dense(
    const float* __restrict__ zr, const float* __restrict__ zi,
    const float* __restrict__ W1, const float* __restrict__ b1,
    const float* __restrict__ W2, const float* __restrict__ b2,
    float* __restrict__ hr, float* __restrict__ hi) {
  __shared__ float sW1p[HID * HID];        // pair layout
  __shared__ float sW2p[HID * HID];
  __shared__ float tiles[2][2][16 * TPAD]; // [wave][A/B]
  const int tid = threadIdx.x;
  for (int it = 0; it < (HID * HID) / 64; ++it) {
    int flat = it * 64 + tid;
    int j = flat >> 6;
    int k = flat & 63;
    const int pidx = (k >> 1) * (2 * HID) + 2 * j + (k & 1);
    sW1p[pidx] = W1[flat];
    sW2p[pidx] = W2[flat];
  }
  __syncthreads();

  const int lane = tid & 31;
  const int wave = tid >> 5;
  const int half = lane >> 4;
  const int l    = lane & 15;
  const int tile = blockIdx.x * 2 + wave;
  if (tile >= NTILES) return;              // wave-uniform guard
  const int row0 = tile * 16;

  float* tA = tiles[wave][0];  // zr, later zr_new
  float* tB = tiles[wave][1];  // zi

  // stage zr / zi tiles, coalesced float4 global reads
  for (int it = 0; it < 8; ++it) {
    int flat = it * 32 + lane;
    int r  = flat >> 4;
    int c4 = (flat & 15) * 4;
    v4f a = *(const v4f*)(zr + (size_t)(row0 + r) * HID + c4);
    v4f b = *(const v4f*)(zi + (size_t)(row0 + r) * HID + c4);
    tA[r * TPAD + c4 + 0] = a.x; tA[r * TPAD + c4 + 1] = a.y;
    tA[r * TPAD + c4 + 2] = a.z; tA[r * TPAD + c4 + 3] = a.w;
    tB[r * TPAD + c4 + 0] = b.x; tB[r * TPAD + c4 + 1] = b.y;
    tB[r * TPAD + c4 + 2] = b.z; tB[r * TPAD + c4 + 3] = b.w;
  }

  const int colp = 2 * l;  // paired-layout column offset base (per jt: +32*jt)

  // ---- phase 1: P = zr@W1^T+b1, Q = zi@W2^T+b2 ----
  v8f P[4], Q[4];
  for (int jt = 0; jt < 4; ++jt) {
    float bv1 = b1[jt * 16 + l];
    float bv2 = b2[jt * 16 + l];
    for (int i = 0; i < 8; ++i) { P[jt][i] = bv1; Q[jt][i] = bv2; }
  }
  for (int k0 = 0; k0 < HID; k0 += 4) {
    const int ka = k0 + 2 * half;
    v2f ar, ai;
    ar.x = tA[l * TPAD + ka]; ar.y = tA[l * TPAD + ka + 1];
    ai.x = tB[l * TPAD + ka]; ai.y = tB[l * TPAD + ka + 1];
    const float* w1row = sW1p + ((k0 >> 1) + half) * (2 * HID);
    const float* w2row = sW2p + ((k0 >> 1) + half) * (2 * HID);
    for (int jt = 0; jt < 4; ++jt) {
      v2f b1f = *(const v2f*)(w1row + 32 * jt + colp);
      v2f b2f = *(const v2f*)(w2row + 32 * jt + colp);
      P[jt] = wmma_f32_16x16x4(ar, b1f, P[jt]);
      Q[jt] = wmma_f32_16x16x4(ai, b2f, Q[jt]);
    }
  }
  // zr_new = P - Q : store pre-relu into tA (A-layout source), relu -> hr
  for (int jt = 0; jt < 4; ++jt)
    for (int i = 0; i < 8; ++i) {
      float v = P[jt][i] - Q[jt][i];
      tA[(i + 8 * half) * TPAD + jt * 16 + l] = v;
      hr[(size_t)(row0 + i + 8 * half) * HID + jt * 16 + l] = v > 0.f ? v : 0.f;
    }

  // ---- phase 2: R = zr_new@W2^T+b2, S = zi@W1^T+b1 ----
  v8f R[4], S[4];
  for (int jt = 0; jt < 4; ++jt) {
    float bv1 = b1[jt * 16 + l];
    float bv2 = b2[jt * 16 + l];
    for (int i = 0; i < 8; ++i) { R[jt][i] = bv2; S[jt][i] = bv1; }
  }
  for (int k0 = 0; k0 < HID; k0 += 4) {
    const int ka = k0 + 2 * half;
    v2f an, ai;
    an.x = tA[l * TPAD + ka]; an.y = tA[l * TPAD + ka + 1];
    ai.x = tB[l * TPAD + ka]; ai.y = tB[l * TPAD + ka + 1];
    const float* w1row = sW1p + ((k0 >> 1) + half) * (2 * HID);
    const float* w2row = sW2p + ((k0 >> 1) + half) * (2 * HID);
    for (int jt = 0; jt < 4; ++jt) {
      v2f b2f = *(const v2f*)(w2row + 32 * jt + colp);
      v2f b1f = *(const v2f*)(w1row + 32 * jt + colp);
      R[jt] = wmma_f32_16x16x4(an, b2f, R[jt]);
      S[jt] = wmma_f32_16x16x4(ai, b1f, S[jt]);
    }
  }
  for (int jt = 0; jt < 4; ++jt)
    for (int i = 0; i < 8; ++i) {
      float v = R[jt][i] + S[jt][i];
      hi[(size_t)(row0 + i + 8 * half) * HID + jt * 16 + l] = v > 0.f ? v : 0.f;
    }
}

// ---------------------------------------------------------------------------
// Kernel 5: output head + log_softmax.  out = logsoftmax([hr|hi]@t2W^T + b)
// Thread per node; 8x128 weights in LDS.
// ---------------------------------------------------------------------------
__global__ __launch_bounds__(256) void k_head(
    const float* __restrict__ hr, const float* __restrict__ hi,
    const float* __restrict__ W, const float* __restrict__ b,
    float* __restrict__ out) {
  __shared__ float sW[OUTD * IN_DIM];
  __shared__ float sb[OUTD];
  for (int i = threadIdx.x; i < OUTD * IN_DIM; i += blockDim.x) sW[i] = W[i];
  if (threadIdx.x < OUTD) sb[threadIdx.x] = b[threadIdx.x];
  __syncthreads();

  const int n = blockIdx.x * blockDim.x + threadIdx.x;
  if (n >= N_NODES) return;
  float acc[OUTD];
  for (int o = 0; o < OUTD; ++o) acc[o] = sb[o];
  const float* xr = hr + (size_t)n * HID;
  const float* xi = hi + (size_t)n * HID;
  for (int k = 0; k < HID; k += 4) {
    v4f a = *(const v4f*)(xr + k);
    for (int o = 0; o < OUTD; ++o) {
      const float* w = sW + o * IN_DIM + k;
      acc[o] += a.x * w[0] + a.y * w[1] + a.z * w[2] + a.w * w[3];
    }
  }
  for (int k = 0; k < HID; k += 4) {
    v4f a = *(const v4f*)(xi + k);
    for (int o = 0; o < OUTD; ++o) {
      const float* w = sW + o * IN_DIM + HID + k;
      acc[o] += a.x * w[0] + a.y * w[1] + a.z * w[2] + a.w * w[3];
    }
  }
  float m = acc[0];
  for (int o = 1; o < OUTD; ++o) m = fmaxf(m, acc[o]);
  float s = 0.f;
  for (int o = 0; o < OUTD; ++o) s += expf(acc[o] - m);
  const float lse = logf(s) + m;
  for (int o = 0; o < OUTD; ++o) out[(size_t)n * OUTD + o] = acc[o] - lse;
}

// ---------------------------------------------------------------------------
extern "C" void kernel_launch(void* const* d_in, const int* in_sizes, int n_in,
                              void* d_out, int out_size, void* d_ws,
                              size_t ws_size, hipStream_t stream) {
  (void)in_sizes; (void)n_in; (void)out_size; (void)ws_size;
  const float* h    = (const float*)d_in[0];
  const float* deg  = (const float*)d_in[1];
  const float* wre  = (const float*)d_in[2];
  const float* wim  = (const float*)d_in[3];
  const float* t1rW = (const float*)d_in[4];
  const float* t1rb = (const float*)d_in[5];
  const float* t1iW = (const float*)d_in[6];
  const float* t1ib = (const float*)d_in[7];
  const float* W1   = (const float*)d_in[8];
  const float* b1   = (const float*)d_in[9];
  const float* W2   = (const float*)d_in[10];
  const float* b2   = (const float*)d_in[11];
  const float* t2W  = (const float*)d_in[12];
  const float* t2b  = (const float*)d_in[13];
  const int*   src  = (const int*)d_in[14];
  const int*   dst  = (const int*)d_in[15];
  float* out = (float*)d_out;

  float* hr = (float*)d_ws;
  float* hi = hr + (size_t)N_NODES * HID;
  float* zr = hi + (size_t)N_NODES * HID;
  float* zi = zr + (size_t)N_NODES * HID;

  k_transform<<<(NTILES + 3) / 4, 128, 0, stream>>>(h, t1rW, t1rb, hr);
  k_transform<<<(NTILES + 3) / 4, 128, 0, stream>>>(h, t1iW, t1ib, hi);

  for (int layer = 0; layer < 2; ++layer) {
    k_zero<<<1024, 256, 0, stream>>>(zr, N_NODES * HID);
    k_zero<<<1024, 256, 0, stream>>>(zi, N_NODES * HID);
    k_agg<<<(N_EDGES + 7) / 8, 256, 0, stream>>>(hr, hi, deg, wre, wim, src,
                                                 dst, zr, zi);
    k_dense<<<(NTILES + 1) / 2, 64, 0, stream>>>(
        zr, zi, W1 + (size_t)layer * HID * HID, b1 + (size_t)layer * HID,
        W2 + (size_t)layer * HID * HID, b2 + (size_t)layer * HID, hr, hi);
  }
  k_head<<<(N_NODES + 255) / 256, 256, 0, stream>>>(hr, hi, t2W, t2b, out);
}